// DGG_LearnableK_debug_10617159156346
// MI455X (gfx1250) — compile-verified
//
#include <hip/hip_runtime.h>

typedef __attribute__((ext_vector_type(2))) float v2f;
typedef __attribute__((ext_vector_type(8))) float v8f;

#define NN 2048
#define IN_DIM 32
#define LATENT 64

// ---------------------------------------------------------------------------
// Kernel 1: z = lrelu(x @ W_ne + b_ne)   [2048,32] x [32,64] -> [2048,64]
// One wave per 16x16 tile; V_WMMA_F32_16X16X4_F32, K loop step 4.
// A frag (16x4 f32): lane m = lane&15, K = 2*(lane>>4) + vgpr
// B frag (4x16 f32): lane n = lane&15, K = 2*(lane>>4) + vgpr
// C/D (16x16 f32):   row = r + 8*(lane>>4), col = lane&15
// ---------------------------------------------------------------------------
__global__ __launch_bounds__(32)
void gemm_z_kernel(const float* __restrict__ x, const float* __restrict__ Wne,
                   const float* __restrict__ bne, float* __restrict__ z)
{
    const int tile = blockIdx.x;
    const int m0 = (tile >> 2) * 16;     // 128 row tiles
    const int n0 = (tile & 3) * 16;      // 4 col tiles
    const int lane = threadIdx.x;
    const int row = lane & 15;
    const int grp = lane >> 4;

    v8f acc = {};
    #pragma unroll
    for (int k = 0; k < IN_DIM; k += 4) {
        const int ka = k + 2 * grp;
        v2f a, b;
        a.x = x[(m0 + row) * IN_DIM + ka];
        a.y = x[(m0 + row) * IN_DIM + ka + 1];
        b.x = Wne[(ka    ) * LATENT + n0 + row];
        b.y = Wne[(ka + 1) * LATENT + n0 + row];
        acc = __builtin_amdgcn_wmma_f32_16x16x4_f32(
            false, a, false, b, (short)0, acc, false, false);
    }

    const float bias = bne[n0 + row];
    #pragma unroll
    for (int r = 0; r < 8; ++r) {
        const int m = m0 + r + 8 * grp;
        float v = acc[r] + bias;
        v = fmaxf(v, 0.01f * v);               // leaky relu
        z[m * LATENT + n0 + row] = v;
    }
}

// ---------------------------------------------------------------------------
// Kernel 2: zi = z @ W_e1a + b_e1 ; zj = z @ W_e1b  (A fragment shared)
// ---------------------------------------------------------------------------
__global__ __launch_bounds__(32)
void gemm_zij_kernel(const float* __restrict__ z, const float* __restrict__ Wa,
                     const float* __restrict__ Wb, const float* __restrict__ be1,
                     float* __restrict__ zi, float* __restrict__ zj)
{
    const int tile = blockIdx.x;
    const int m0 = (tile >> 2) * 16;
    const int n0 = (tile & 3) * 16;
    const int lane = threadIdx.x;
    const int row = lane & 15;
    const int grp = lane >> 4;

    v8f acca = {};
    v8f accb = {};
    #pragma unroll
    for (int k = 0; k < LATENT; k += 4) {
        const int ka = k + 2 * grp;
        v2f a, ba, bb;
        a.x = z[(m0 + row) * LATENT + ka];
        a.y = z[(m0 + row) * LATENT + ka + 1];
        ba.x = Wa[(ka    ) * LATENT + n0 + row];
        ba.y = Wa[(ka + 1) * LATENT + n0 + row];
        bb.x = Wb[(ka    ) * LATENT + n0 + row];
        bb.y = Wb[(ka + 1) * LATENT + n0 + row];
        acca = __builtin_amdgcn_wmma_f32_16x16x4_f32(
            false, a, false, ba, (short)0, acca, false, false);
        accb = __builtin_amdgcn_wmma_f32_16x16x4_f32(
            false, a, false, bb, (short)0, accb, false, false);
    }

    const float bias = be1[n0 + row];
    #pragma unroll
    for (int r = 0; r < 8; ++r) {
        const int m = m0 + r + 8 * grp;
        zi[m * LATENT + n0 + row] = acca[r] + bias;   // fold b_e1 into zi
        zj[m * LATENT + n0 + row] = accb[r];
    }
}

// ---------------------------------------------------------------------------
// Kernel 3: pairwise stage, register-blocked.
// 256 threads compute a 64x64 (i,j) tile; each thread owns a 4x4 sub-tile
// (i stride 16, j stride 16).  l-dimension vectorized by 4 via ds_load_b128.
// DS traffic: 8 b128 loads serve 16 outputs x 4 l-steps (0.5 dword/output/l,
// 4x less than the naive scheme) -> VALU-bound inner loop.
// sj stride 68 floats = 272 B: 16-B aligned, bank = (4*tx + l) % 64 ->
// conflict-free across the 16 tx lanes (lanes 16-31 broadcast-duplicate).
// ---------------------------------------------------------------------------
__global__ __launch_bounds__(256)
void edge_kernel(const float* __restrict__ zi, const float* __restrict__ zj,
                 const float* __restrict__ w, const float* __restrict__ be2,
                 const float* __restrict__ gum, float* __restrict__ out)
{
    __shared__ float si[64 * 64];
    __shared__ float sj[64 * 68];
    __shared__ float sw[64];

    const int tid = threadIdx.x;
    const int tx = tid & 15;
    const int ty = tid >> 4;
    const int i0 = blockIdx.y * 64;
    const int j0 = blockIdx.x * 64;

    // Stage tiles: 64 rows x 16 float4 per tile, 4 float4 per thread per tile.
    #pragma unroll
    for (int e = tid; e < 64 * 16; e += 256) {
        const int r = e >> 4;
        const int c = (e & 15) * 4;
        *(float4*)&si[r * 64 + c] = *(const float4*)&zi[(i0 + r) * LATENT + c];
        *(float4*)&sj[r * 68 + c] = *(const float4*)&zj[(j0 + r) * LATENT + c];
    }
    if (tid < 64) sw[tid] = w[tid];
    __syncthreads();

    float acc[4][4] = {};
    for (int l = 0; l < LATENT; l += 4) {
        const float4 wv = *(const float4*)&sw[l];
        float4 va[4], ub[4];
        #pragma unroll
        for (int a = 0; a < 4; ++a)
            va[a] = *(const float4*)&si[(ty + 16 * a) * 64 + l];
        #pragma unroll
        for (int b = 0; b < 4; ++b)
            ub[b] = *(const float4*)&sj[(tx + 16 * b) * 68 + l];

        #pragma unroll
        for (int a = 0; a < 4; ++a) {
            #pragma unroll
            for (int b = 0; b < 4; ++b) {
                float v0 = va[a].x + ub[b].x;  v0 = fmaxf(v0, 0.01f * v0);
                float v1 = va[a].y + ub[b].y;  v1 = fmaxf(v1, 0.01f * v1);
                float v2 = va[a].z + ub[b].z;  v2 = fmaxf(v2, 0.01f * v2);
                float v3 = va[a].w + ub[b].w;  v3 = fmaxf(v3, 0.01f * v3);
                float s = acc[a][b];
                s = fmaf(v0, wv.x, s);
                s = fmaf(v1, wv.y, s);
                s = fmaf(v2, wv.z, s);
                s = fmaf(v3, wv.w, s);
                acc[a][b] = s;
            }
        }
    }

    const float bb = be2[0];
    #pragma unroll
    for (int a = 0; a < 4; ++a) {
        #pragma unroll
        for (int b = 0; b < 4; ++b) {
            const int i = i0 + ty + 16 * a;
            const int j = j0 + tx + 16 * b;
            const float t = acc[a][b] + bb;
            const float p = 1.0f / (1.0f + __expf(-t));
            // exp(log(p + 1e-8) + g) == (p + 1e-8) * exp(g)
            out[i * NN + j] = (p + 1e-8f) * __expf(gum[i * NN + j]);
        }
    }
}

// ---------------------------------------------------------------------------
extern "C" void kernel_launch(void* const* d_in, const int* in_sizes, int n_in,
                              void* d_out, int out_size, void* d_ws, size_t ws_size,
                              hipStream_t stream)
{
    const float* x    = (const float*)d_in[0];
    // d_in[1] = in_adj (unused by the reference output)
    const float* gum  = (const float*)d_in[2];
    const float* Wne  = (const float*)d_in[3];
    const float* bne  = (const float*)d_in[4];
    const float* We1a = (const float*)d_in[5];
    const float* We1b = (const float*)d_in[6];
    const float* be1  = (const float*)d_in[7];
    const float* we2  = (const float*)d_in[8];
    const float* be2  = (const float*)d_in[9];
    float* out = (float*)d_out;

    float* z  = (float*)d_ws;            // 2048*64 f32
    float* zi = z  + NN * LATENT;        // 2048*64 f32 (b_e1 folded in)
    float* zj = zi + NN * LATENT;        // 2048*64 f32

    const int gemm_tiles = (NN / 16) * (LATENT / 16);   // 512 waves
    gemm_z_kernel  <<<gemm_tiles, 32, 0, stream>>>(x, Wne, bne, z);
    gemm_zij_kernel<<<gemm_tiles, 32, 0, stream>>>(z, We1a, We1b, be1, zi, zj);

    dim3 grid(NN / 64, NN / 64);         // 32 x 32 tiles of 64x64
    edge_kernel<<<grid, 256, 0, stream>>>(zi, zj, we2, be2, gum, out);
}